// DensityGenerator_57140244906317
// MI455X (gfx1250) — compile-verified
//
#include <hip/hip_runtime.h>

// Shapes fixed by setup_inputs(): maps (B=32, NH=16, 256, 256) f32, weight (1,1,3,3) f32.
constexpr int NH = 16;
constexpr int H = 256;
constexpr int W = 256;
constexpr int HW = H * W;              // 65536
constexpr int PER_SAMPLE = NH * HW;    // 1048576
constexpr int P = 64;                  // partial blocks per sample in k1
constexpr float TEMPERATURE = 2.0f;

__device__ __forceinline__ float wave_max(float v) {
#pragma unroll
  for (int o = 16; o > 0; o >>= 1) v = fmaxf(v, __shfl_xor(v, o, 32));
  return v;
}
__device__ __forceinline__ float wave_sum(float v) {
#pragma unroll
  for (int o = 16; o > 0; o >>= 1) v += __shfl_xor(v, o, 32);
  return v;
}

// ---------------- Kernel 1: per-sample partial max / sum ----------------
// grid = B*P blocks, 256 threads. Each block reduces PER_SAMPLE/P elements.
__global__ void k1_partial_stats(const float* __restrict__ maps,
                                 float* __restrict__ pmax,
                                 float* __restrict__ psum) {
  const int blk = blockIdx.x;
  const int s = blk / P;
  const int chunk = blk - s * P;
  const int tid = threadIdx.x;  // 0..255
  const float4* in4 = (const float4*)maps;
  const size_t base = (size_t)s * (PER_SAMPLE / 4) +
                      (size_t)chunk * (PER_SAMPLE / 4 / P) + tid;
  float lmax = -3.402823466e38f, lsum = 0.f;
#pragma unroll
  for (int j = 0; j < PER_SAMPLE / 4 / P / 256; ++j) {  // 16 iters
    float4 v = in4[base + j * 256];
    lmax = fmaxf(lmax, fmaxf(fmaxf(v.x, v.y), fmaxf(v.z, v.w)));
    lsum += (v.x + v.y) + (v.z + v.w);
  }
  lmax = wave_max(lmax);
  lsum = wave_sum(lsum);
  __shared__ float smx[8], ssm[8];
  const int wave = tid >> 5, lane = tid & 31;
  if (lane == 0) { smx[wave] = lmax; ssm[wave] = lsum; }
  __syncthreads();
  if (tid == 0) {
    float m = smx[0], s2 = ssm[0];
#pragma unroll
    for (int i = 1; i < 8; ++i) { m = fmaxf(m, smx[i]); s2 += ssm[i]; }
    pmax[blk] = m;
    psum[blk] = s2;
  }
}

// ---------------- Kernel 2: finalize per-sample max / mean ----------------
// grid = B blocks, one wave (32 threads) each; P == 64 -> 2 partials per lane.
__global__ void k2_finalize(const float* __restrict__ pmax,
                            const float* __restrict__ psum,
                            float* __restrict__ smx,
                            float* __restrict__ smean) {
  const int s = blockIdx.x, t = threadIdx.x;  // t in [0,32)
  float m = fmaxf(pmax[s * P + t], pmax[s * P + t + 32]);
  float sm = psum[s * P + t] + psum[s * P + t + 32];
  m = wave_max(m);
  sm = wave_sum(sm);
  if (t == 0) {
    smx[s] = m;
    smean[s] = sm * (1.0f / (float)PER_SAMPLE);
  }
}

// ------- Kernel 3: outlier replace (==max -> mean) + mean over heads -------
// grid = (HW/4/256, B), 256 threads; each thread owns one float4 pixel group.
__global__ void k3_headmean(const float* __restrict__ maps,
                            const float* __restrict__ smx,
                            const float* __restrict__ smean,
                            float* __restrict__ mbuf) {
  const int s = blockIdx.y;
  const int q = blockIdx.x * blockDim.x + threadIdx.x;  // [0, HW/4)
  const float4* in4 = (const float4*)maps;
  float4* m4 = (float4*)mbuf;
  const float mx = smx[s], mn = smean[s];
  float4 acc = {0.f, 0.f, 0.f, 0.f};
  const size_t base = (size_t)s * (PER_SAMPLE / 4) + q;
#pragma unroll
  for (int nh = 0; nh < NH; ++nh) {
    float4 v = in4[base + (size_t)nh * (HW / 4)];
    acc.x += (v.x == mx) ? mn : v.x;
    acc.y += (v.y == mx) ? mn : v.y;
    acc.z += (v.z == mx) ? mn : v.z;
    acc.w += (v.w == mx) ? mn : v.w;
  }
  const float sc = 1.0f / (float)NH;
  acc.x *= sc; acc.y *= sc; acc.z *= sc; acc.w *= sc;
  m4[(size_t)s * (HW / 4) + q] = acc;
}

// ------ Kernel 4: async-stage m slice to LDS, 3x3 conv, fused softmax ------
// One workgroup per sample. 1024 threads = 32 waves. Dynamic LDS:
//   tile[HW] (256 KiB, async-staged) + red[64] reduction scratch.
// Each thread owns a 64-row column segment: col = tid&255, rows [rb*64, rb*64+64).
// A sliding 3-row register window gives 3 LDS loads per output pixel.
__global__ __launch_bounds__(1024) void k4_conv_softmax(
    const float* __restrict__ mbuf, const float* __restrict__ cw,
    float* __restrict__ out) {
  extern __shared__ float lds[];
  float* tile = lds;       // HW floats
  float* red = lds + HW;   // reduction scratch
  const int b = blockIdx.x;
  const int tid = threadIdx.x;                 // 0..1023
  const int lane = tid & 31, wave = tid >> 5;  // 32 waves

  // 3x3 weights (all 1/9 for the box kernel, but honor the input buffer).
  float wgt[9];
#pragma unroll
  for (int i = 0; i < 9; ++i) wgt[i] = cw[i];

  // ---- Async global -> LDS staging of the whole 256x256 slice (256 KiB).
  // 16 iterations x 1024 lanes x 16 B. Generic LDS addr truncates to the
  // 32-bit LDS offset expected by global_load_async_to_lds (ASYNCcnt path).
  const float* src = mbuf + (size_t)b * HW;
  const unsigned lds_base = (unsigned)(uintptr_t)tile;
#pragma unroll
  for (int it = 0; it < (HW * 4) / (1024 * 16); ++it) {  // 16 iters
    unsigned off = (unsigned)(tid + it * 1024) * 16u;    // byte offset
    unsigned la = lds_base + off;
    asm volatile("global_load_async_to_lds_b128 %0, %1, %2"
                 :: "v"(la), "v"(off), "s"(src)
                 : "memory");
  }
  asm volatile("s_wait_asynccnt 0x0" ::: "memory");
  __syncthreads();

  // ---- 3x3 conv (zero pad) / temperature; sliding 3-row register window.
  constexpr int RT = 64;            // rows per thread
  const int c = tid & (W - 1);      // column owned by this thread
  const int r0 = (tid >> 8) * RT;   // first row of this thread's segment
  const bool cl = (c > 0), cr = (c < W - 1);

  float a0[3], a1[3], a2[3];
  // row r0-1 (zero if out of image)
  if (r0 > 0) {
    const int ro = (r0 - 1) << 8;
    a0[0] = cl ? tile[ro + c - 1] : 0.f;
    a0[1] = tile[ro + c];
    a0[2] = cr ? tile[ro + c + 1] : 0.f;
  } else {
    a0[0] = a0[1] = a0[2] = 0.f;
  }
  // row r0
  {
    const int ro = r0 << 8;
    a1[0] = cl ? tile[ro + c - 1] : 0.f;
    a1[1] = tile[ro + c];
    a1[2] = cr ? tile[ro + c + 1] : 0.f;
  }

  float xv[RT];
  float lmax = -3.402823466e38f;
#pragma unroll
  for (int k = 0; k < RT; ++k) {
    const int r = r0 + k;
    // incoming row r+1 (zero if past bottom edge)
    if (r + 1 < H) {
      const int ro = (r + 1) << 8;
      a2[0] = cl ? tile[ro + c - 1] : 0.f;
      a2[1] = tile[ro + c];
      a2[2] = cr ? tile[ro + c + 1] : 0.f;
    } else {
      a2[0] = a2[1] = a2[2] = 0.f;
    }
    float acc = wgt[0] * a0[0] + wgt[1] * a0[1] + wgt[2] * a0[2] +
                wgt[3] * a1[0] + wgt[4] * a1[1] + wgt[5] * a1[2] +
                wgt[6] * a2[0] + wgt[7] * a2[1] + wgt[8] * a2[2];
    const float xt = acc * (1.0f / TEMPERATURE);
    xv[k] = xt;
    lmax = fmaxf(lmax, xt);
    // shift window (register renames after full unroll)
    a0[0] = a1[0]; a0[1] = a1[1]; a0[2] = a1[2];
    a1[0] = a2[0]; a1[1] = a2[1]; a1[2] = a2[2];
  }

  // ---- block-wide max (wave32 shuffles + 32-entry LDS tree)
  lmax = wave_max(lmax);
  if (lane == 0) red[wave] = lmax;
  __syncthreads();
  if (wave == 0) {
    float v = red[lane];
    v = wave_max(v);
    if (lane == 0) red[32] = v;
  }
  __syncthreads();
  const float gmax = red[32];
  __syncthreads();

  // ---- exp + block-wide sum
  float lsum = 0.f;
#pragma unroll
  for (int k = 0; k < RT; ++k) {
    const float e = __expf(xv[k] - gmax);
    xv[k] = e;
    lsum += e;
  }
  lsum = wave_sum(lsum);
  if (lane == 0) red[wave] = lsum;
  __syncthreads();
  if (wave == 0) {
    float v = red[lane];
    v = wave_sum(v);
    if (lane == 0) red[32] = v;
  }
  __syncthreads();
  const float inv = 1.0f / red[32];

  // ---- normalized write-out (lanes span consecutive columns -> coalesced)
  float* dst = out + (size_t)b * HW;
#pragma unroll
  for (int k = 0; k < RT; ++k) dst[((r0 + k) << 8) + c] = xv[k] * inv;
}

// ------------------------------- launcher ---------------------------------
extern "C" void kernel_launch(void* const* d_in, const int* in_sizes, int n_in,
                              void* d_out, int out_size, void* d_ws,
                              size_t ws_size, hipStream_t stream) {
  (void)n_in; (void)out_size; (void)ws_size;
  const float* maps = (const float*)d_in[0];
  const float* cw = (const float*)d_in[1];
  float* out = (float*)d_out;
  const int B = in_sizes[0] / PER_SAMPLE;  // 32

  // Workspace layout (floats): pmax[B*P] | psum[B*P] | smx[B] | smean[B] | m[B*HW]
  float* ws = (float*)d_ws;
  float* pmax = ws;
  float* psum = pmax + (size_t)B * P;
  float* smx = psum + (size_t)B * P;
  float* smean = smx + B;
  size_t moff = ((size_t)(2 * B * P + 2 * B) + 3) & ~(size_t)3;  // float4 align
  float* mbuf = ws + moff;

  k1_partial_stats<<<dim3(B * P), dim3(256), 0, stream>>>(maps, pmax, psum);
  k2_finalize<<<dim3(B), dim3(32), 0, stream>>>(pmax, psum, smx, smean);
  k3_headmean<<<dim3(HW / 4 / 256, B), dim3(256), 0, stream>>>(maps, smx,
                                                               smean, mbuf);
  const size_t shmem = (size_t)(HW + 64) * sizeof(float);  // ~256.25 KiB
  k4_conv_softmax<<<dim3(B), dim3(1024), shmem, stream>>>(mbuf, cw, out);
}